// NormMixingOutput_14448269983913
// MI455X (gfx1250) — compile-verified
//
#include <hip/hip_runtime.h>
#include <hip/hip_bf16.h>
#include <math.h>

typedef __attribute__((ext_vector_type(2))) float v2f;
typedef __attribute__((ext_vector_type(8))) float v8f;
typedef int v4i_ __attribute__((vector_size(16)));   // matches async-LDS builtin param

#define B_  2
#define L_  256
#define H_  512
#define HH_ 8
#define DH_ 64

// detect gfx1250 async-to-LDS builtins (fallback: plain LDS staging)
#if defined(__has_builtin)
#if __has_builtin(__builtin_amdgcn_global_load_async_to_lds_b128) && \
    __has_builtin(__builtin_amdgcn_s_wait_asynccnt)
#define USE_ASYNC_LDS 1
#endif
#endif
#ifndef USE_ASYNC_LDS
#define USE_ASYNC_LDS 0
#endif

// workspace layout (float offsets)
#define WS_T      0
#define WS_TA     (WS_T + B_*HH_*L_*H_)       // per-(b,h,j) sum_v T
#define WS_TB     (WS_TA + B_*HH_*L_)         // per-(b,h,j) sum_v T*w^2
#define WS_CROSS  (WS_TB + B_*HH_*L_)         // per-(b,h,j) sum_v T*hid*w^2
#define WS_G      (WS_CROSS + B_*HH_*L_)      // per-(b,j) 8x8 Gram (w^2-weighted)
#define WS_SIGMA  (WS_G + B_*L_*64)
#define WS_HA     (WS_SIGMA + B_*L_)          // sum_v hid
#define WS_HB     (WS_HA + B_*L_)             // sum_v hid*w^2
#define WS_HH2    (WS_HB + B_*L_)             // sum_v hid^2*w^2
#define WS_AO     (WS_HH2 + B_*L_)            // attn_out [B,L,H]
#define WS_W2     (WS_AO + B_*L_*H_)          // sum_v w^2 (scalar)

// output layout (floats)
#define OUT_NORM   0         // [B,L,L]
#define OUT_ATTN   131072    // [B,L]
#define OUT_ATTNR  131584    // [B,L]
#define OUT_POST   132096    // [B,L]
#define OUT_ENT    132608    // [B]

__device__ __forceinline__ float wave_sum(float v) {
#pragma unroll
  for (int off = 16; off > 0; off >>= 1) v += __shfl_xor(v, off, 32);
  return v;
}

// blockDim.x == 256 (8 wave32s)
__device__ __forceinline__ float block_sum_256(float v, float* s8) {
  int tid = threadIdx.x;
  float w = wave_sum(v);
  __syncthreads();
  if ((tid & 31) == 0) s8[tid >> 5] = w;
  __syncthreads();
  return s8[0]+s8[1]+s8[2]+s8[3]+s8[4]+s8[5]+s8[6]+s8[7];
}

// ---------------------------------------------------------------------------
// K0: zero entropy accumulators
__global__ void k0_init(float* __restrict__ out) {
  if (threadIdx.x < B_) out[OUT_ENT + threadIdx.x] = 0.0f;
}

// ---------------------------------------------------------------------------
// K1: T[b,h,j,v] = sum_d V[b,h,j,d] * W[v*H + h*DH + d]   (f32 WMMA 16x16x4)
// one 16x16 output tile per wave; 2 waves per block
__global__ void k1_transform_wmma(const float* __restrict__ V,
                                  const float* __restrict__ W,
                                  float* __restrict__ T) {
  int lane = threadIdx.x & 31;
  int tile = blockIdx.x * 2 + (threadIdx.x >> 5);
  int tv = tile & 31; tile >>= 5;
  int tj = tile & 15; tile >>= 4;
  int h  = tile & 7;  int b = tile >> 3;

  const float* Ap = V + ((b*HH_ + h)*L_ + tj*16) * DH_;  // [16 x 64], stride DH_
  const float* Bp = W + h*DH_;                           // B[d,v] = W[v*H_ + h*DH_ + d]
  int row = lane & 15;
  int kk  = (lane >> 4) << 1;   // lanes 0-15: K=k0+0/1 ; lanes 16-31: K=k0+2/3

  v8f acc = {};
#pragma unroll
  for (int k0 = 0; k0 < DH_; k0 += 4) {
    v2f a, bq;
    a.x  = Ap[row*DH_ + k0 + kk];
    a.y  = Ap[row*DH_ + k0 + kk + 1];
    bq.x = Bp[(tv*16 + row)*H_ + k0 + kk];
    bq.y = Bp[(tv*16 + row)*H_ + k0 + kk + 1];
    acc = __builtin_amdgcn_wmma_f32_16x16x4_f32(false, a, false, bq,
                                                (short)0, acc, false, false);
  }
  // C/D layout: VGPR p -> M=p (lanes 0-15) / M=p+8 (lanes 16-31), N = lane&15
  float* Tp = T + ((b*HH_ + h)*L_ + tj*16)*H_ + tv*16;
  int mb = (lane >> 4) * 8;
#pragma unroll
  for (int p = 0; p < 8; ++p) Tp[(mb + p)*H_ + row] = acc[p];
}

// ---------------------------------------------------------------------------
// K2: per (b,j) statistics: tA/tB/cross per head, 8x8 Gram, hid sums, sigma, W2
__global__ void k2_stats(const float* __restrict__ T,
                         const float* __restrict__ hid,
                         const float* __restrict__ lnw,
                         const float* __restrict__ pre,
                         float* __restrict__ ws) {
  __shared__ float sT[HH_][H_];
  __shared__ float sw2[H_];
  __shared__ float sh[H_];
  int bj = blockIdx.x;                  // b*L + j
  int b = bj >> 8, j = bj & 255;
  int tid = threadIdx.x;

  for (int idx = tid; idx < HH_*H_; idx += 256) {
    int hh = idx >> 9, v = idx & 511;
    sT[hh][v] = T[((b*HH_ + hh)*L_ + j)*H_ + v];
  }
  for (int v = tid; v < H_; v += 256) {
    float w = lnw[v];
    sw2[v] = w * w;
    sh[v]  = hid[(b*L_ + j)*H_ + v];
  }
  __syncthreads();

  int wv = tid >> 5, lane = tid & 31;

  // per-head linear sums: wave wv handles head wv
  {
    float ta = 0, tb = 0, cr = 0;
    for (int v = lane; v < H_; v += 32) {
      float t = sT[wv][v];
      ta += t;
      tb += t * sw2[v];
      cr += t * sh[v] * sw2[v];
    }
    ta = wave_sum(ta); tb = wave_sum(tb); cr = wave_sum(cr);
    if (lane == 0) {
      ws[WS_TA    + (b*HH_ + wv)*L_ + j] = ta;
      ws[WS_TB    + (b*HH_ + wv)*L_ + j] = tb;
      ws[WS_CROSS + (b*HH_ + wv)*L_ + j] = cr;
    }
  }
  // Gram: wave wv handles row h1=wv
#pragma unroll 1
  for (int h2 = 0; h2 < HH_; ++h2) {
    float g = 0;
    for (int v = lane; v < H_; v += 32) g += sT[wv][v] * sT[h2][v] * sw2[v];
    g = wave_sum(g);
    if (lane == 0) ws[WS_G + bj*64 + wv*8 + h2] = g;
  }
  // extras
  if (wv == 0) {
    float s = 0;
    for (int v = lane; v < H_; v += 32) s += sh[v];
    s = wave_sum(s);
    if (lane == 0) ws[WS_HA + bj] = s;
  } else if (wv == 1) {
    float s = 0;
    for (int v = lane; v < H_; v += 32) s += sh[v] * sw2[v];
    s = wave_sum(s);
    if (lane == 0) ws[WS_HB + bj] = s;
  } else if (wv == 2) {
    float s = 0;
    for (int v = lane; v < H_; v += 32) s += sh[v] * sh[v] * sw2[v];
    s = wave_sum(s);
    if (lane == 0) ws[WS_HH2 + bj] = s;
  } else if (wv == 3) {
    float s1 = 0, s2 = 0;
    for (int v = lane; v < H_; v += 32) {
      float p = pre[(b*L_ + j)*H_ + v];
      s1 += p; s2 += p * p;
    }
    s1 = wave_sum(s1); s2 = wave_sum(s2);
    if (lane == 0) {
      float mean = s1 * (1.0f/H_);
      float var  = s2 * (1.0f/H_) - mean*mean;
      ws[WS_SIGMA + bj] = sqrtf(var + 1e-5f);
    }
  } else if (wv == 4 && bj == 0) {
    float s = 0;
    for (int v = lane; v < H_; v += 32) s += sw2[v];
    s = wave_sum(s);
    if (lane == 0) ws[WS_W2] = s;
  }
}

// ---------------------------------------------------------------------------
// K3: attn_out[b,i,v] = sum_h sum_j P[b,h,i,j] * T[b,h,j,v]
// 8 wave32s per block share one async-LDS-staged 16x256 A (P) tile;
// each wave computes a different 16-column tile via f32 WMMA (K=256 x 8 heads).
__global__ void k3_attnout_wmma(const float* __restrict__ P,
                                const float* __restrict__ T,
                                float* __restrict__ AO) {
  __shared__ float sA[16 * L_];          // 16 KB: one head's P tile
  int tid  = threadIdx.x;
  int lane = tid & 31, wv = tid >> 5;
  int blk  = blockIdx.x;                 // b(2) x ti(16) x tvg(4)
  int tvg = blk & 3; blk >>= 2;
  int ti  = blk & 15; int b = blk >> 4;
  int tv  = tvg * 8 + wv;                // per-wave output column tile
  int row = lane & 15;
  int kk  = (lane >> 4) << 1;

  v8f acc = {};
#pragma unroll 1
  for (int h = 0; h < HH_; ++h) {
    const float* Ap = P + ((b*HH_ + h)*L_ + ti*16)*L_;
    __syncthreads();                     // previous tile fully consumed
#if USE_ASYNC_LDS
    {
      const float* gsrc = Ap + tid * 4;  // 16B per thread per issue
      float* ldst = sA + tid * 4;
#pragma unroll
      for (int it = 0; it < 4; ++it) {
        __builtin_amdgcn_global_load_async_to_lds_b128(
            (v4i_*)gsrc, (v4i_*)ldst, 0, 0);
        gsrc += 1024; ldst += 1024;
      }
      __builtin_amdgcn_s_wait_asynccnt(0);
    }
#else
    for (int idx = tid; idx < 16 * L_; idx += 256) sA[idx] = Ap[idx];
#endif
    if (h + 1 < HH_) {                   // prefetch the 4 P rows after this tile into L2
      __builtin_prefetch(Ap + 16 * L_ + tid * 4, 0, 1);
    }
    __syncthreads();

    const float* Bp = T + ((b*HH_ + h)*L_)*H_ + tv*16;  // B[k,n] = T[...,k, tv*16+n]
#pragma unroll 4
    for (int k0 = 0; k0 < L_; k0 += 4) {
      v2f a, bq;
      a.x  = sA[row*L_ + k0 + kk];
      a.y  = sA[row*L_ + k0 + kk + 1];
      bq.x = Bp[(k0 + kk)*H_ + row];
      bq.y = Bp[(k0 + kk + 1)*H_ + row];
      acc = __builtin_amdgcn_wmma_f32_16x16x4_f32(false, a, false, bq,
                                                  (short)0, acc, false, false);
    }
  }
  float* Op = AO + (b*L_ + ti*16)*H_ + tv*16;
  int mb = (lane >> 4) * 8;
#pragma unroll
  for (int p = 0; p < 8; ++p) Op[(mb + p)*H_ + row] = acc[p];
}

// ---------------------------------------------------------------------------
// K4: mixing ratios per (b,i). mix_sw = attn_out - diag; Σ_j post_ln collapses
// analytically to (w/σ)(S_v - mean(S)) with S = attn_out + hidden.
__global__ void k4_mixing(const float* __restrict__ P,
                          const float* __restrict__ T,
                          const float* __restrict__ hid,
                          const float* __restrict__ lnw,
                          const float* __restrict__ ws,
                          float* __restrict__ out) {
  __shared__ float s8[8];
  int bi = blockIdx.x; int b = bi >> 8, i = bi & 255;
  int tid = threadIdx.x;
  const float* AO = ws + WS_AO;
  float sigma = ws[WS_SIGMA + bi];

  float dv[2], aov[2], hv[2], wv_[2];
#pragma unroll
  for (int t = 0; t < 2; ++t) {
    int v = tid + t*256;
    float d = 0;
#pragma unroll
    for (int h = 0; h < HH_; ++h) {
      float pdiag = P[((b*HH_ + h)*L_ + i)*L_ + i];
      d += pdiag * T[((b*HH_ + h)*L_ + i)*H_ + v];
    }
    dv[t]  = d;
    aov[t] = AO[(b*L_ + i)*H_ + v];
    hv[t]  = hid[(b*L_ + i)*H_ + v];
    wv_[t] = lnw[v];
  }
  float pn_sw = 0, mn_sw = 0, pn_z = 0, sS = 0, szd = 0;
#pragma unroll
  for (int t = 0; t < 2; ++t) {
    float d = dv[t], ao = aov[t], hh = hv[t];
    float mix = ao - d;                 // same for sw and z (residual cancels)
    pn_sw += d*d;
    mn_sw += mix*mix;
    float zd = d + hh;
    pn_z += zd*zd;
    sS  += ao + hh;
    szd += zd;
  }
  pn_sw = block_sum_256(pn_sw, s8);
  mn_sw = block_sum_256(mn_sw, s8);
  pn_z  = block_sum_256(pn_z,  s8);
  sS    = block_sum_256(sS,    s8);
  szd   = block_sum_256(szd,   s8);
  float M  = sS  * (1.0f/H_);
  float md = szd * (1.0f/H_);
  float pp = 0, mp = 0;
#pragma unroll
  for (int t = 0; t < 2; ++t) {
    float zd = dv[t] + hv[t];
    float S  = aov[t] + hv[t];
    float pres     = (zd - md) * wv_[t] / sigma;
    float sum_post = (S  - M)  * wv_[t] / sigma;
    float mix = sum_post - pres;
    pp += pres*pres;
    mp += mix*mix;
  }
  pp = block_sum_256(pp, s8);
  mp = block_sum_256(mp, s8);
  if (tid == 0) {
    float mn = sqrtf(mn_sw);
    out[OUT_ATTN  + bi] = mn / (mn + sqrtf(pn_sw) + 1e-12f);
    out[OUT_ATTNR + bi] = mn / (mn + sqrtf(pn_z)  + 1e-12f);
    float mpn = sqrtf(mp);
    out[OUT_POST  + bi] = mpn / (mpn + sqrtf(pp) + 1e-12f);
  }
}

// ---------------------------------------------------------------------------
// K5: entropy mean per batch
__global__ void k5_entropy(const float* __restrict__ P, float* __restrict__ out) {
  __shared__ float s8[8];
  int blk = blockIdx.x;
  int i = blk & 255; int h = (blk >> 8) & 7; int b = blk >> 11;
  int j = threadIdx.x;
  float p = P[((b*HH_ + h)*L_ + i)*L_ + j];
  float e = -p * logf(p + 1e-12f);
  e = block_sum_256(e, s8);
  if (j == 0) {
    float denom = logf((float)(i + 1)) + 1e-12f;
    atomicAdd(&out[OUT_ENT + b], (e / denom) * (1.0f/(HH_*L_)));
  }
}

// ---------------------------------------------------------------------------
// K6: post_ln_norm[b,i,j] = sqrt(pGp - 2m*pB + m^2*W2)/sigma_i, m = pA/H
__global__ void k6_postlnnorm(const float* __restrict__ P,
                              const float* __restrict__ ws,
                              float* __restrict__ out) {
  int bi = blockIdx.x; int b = bi >> 8, i = bi & 255;
  int j = threadIdx.x;
  float W2 = ws[WS_W2];
  float sigma = ws[WS_SIGMA + bi];
  float p[HH_];
  float A = 0, Bs = 0;
#pragma unroll
  for (int h = 0; h < HH_; ++h) {
    p[h] = P[((b*HH_ + h)*L_ + i)*L_ + j];
    A  += p[h] * ws[WS_TA + (b*HH_ + h)*L_ + j];
    Bs += p[h] * ws[WS_TB + (b*HH_ + h)*L_ + j];
  }
  const float* Gp = ws + WS_G + (b*L_ + j)*64;
  float Cs = 0;
#pragma unroll
  for (int h2 = 0; h2 < 8; ++h2) {
    float q = 0;
#pragma unroll
    for (int h1 = 0; h1 < 8; ++h1) q += p[h1] * Gp[h1*8 + h2];
    Cs += q * p[h2];
  }
  if (j == i) {   // residual on the diagonal
    int bj = b*L_ + j;
    A  += ws[WS_HA + bj];
    Bs += ws[WS_HB + bj];
    float cx = 0;
#pragma unroll
    for (int h = 0; h < 8; ++h) cx += p[h] * ws[WS_CROSS + (b*HH_ + h)*L_ + j];
    Cs += 2.0f*cx + ws[WS_HH2 + bj];
  }
  float m = A * (1.0f/H_);
  float nsq = Cs - 2.0f*m*Bs + m*m*W2;
  nsq = fmaxf(nsq, 0.0f);
  out[OUT_NORM + (b*L_ + i)*L_ + j] = sqrtf(nsq) / sigma;
}

// ---------------------------------------------------------------------------
extern "C" void kernel_launch(void* const* d_in, const int* in_sizes, int n_in,
                              void* d_out, int out_size, void* d_ws, size_t ws_size,
                              hipStream_t stream) {
  (void)in_sizes; (void)n_in; (void)out_size; (void)ws_size;
  const float* hid = (const float*)d_in[0];   // hidden_states [B,L,H]
  const float* P   = (const float*)d_in[1];   // attention_probs [B,Hh,L,L]
  const float* V   = (const float*)d_in[2];   // value_layer [B,Hh,L,Dh]
  const float* W   = (const float*)d_in[3];   // out_proj_weight [H,H]
  const float* lnw = (const float*)d_in[4];   // ln_weight [H]
  const float* pre = (const float*)d_in[5];   // pre_ln_states [B,L,H]
  float* out = (float*)d_out;
  float* ws  = (float*)d_ws;
  float* T   = ws + WS_T;
  float* AO  = ws + WS_AO;

  k0_init<<<1, 32, 0, stream>>>(out);
  k1_transform_wmma<<<(B_*HH_*16*32)/2, 64, 0, stream>>>(V, W, T);
  k2_stats<<<B_*L_, 256, 0, stream>>>(T, hid, lnw, pre, ws);
  k3_attnout_wmma<<<B_*16*4, 256, 0, stream>>>(P, T, AO);
  k4_mixing<<<B_*L_, 256, 0, stream>>>(P, T, hid, lnw, ws, out);
  k5_entropy<<<B_*HH_*L_, 256, 0, stream>>>(P, out);
  k6_postlnnorm<<<B_*L_, 256, 0, stream>>>(P, ws, out);
}